// GraphSAGEEncoder_26336739459482
// MI455X (gfx1250) — compile-verified
//
#include <hip/hip_runtime.h>

#define D 128
#define SCAN_BS 512

typedef __attribute__((ext_vector_type(2))) float v2f;
typedef __attribute__((ext_vector_type(8))) float v8f;

// ---------------------------------------------------------------- zero fill
__global__ void gs_zero_i32(int* __restrict__ p, long long n) {
    long long i = (long long)blockIdx.x * blockDim.x + threadIdx.x;
    long long stride = (long long)gridDim.x * blockDim.x;
    for (; i < n; i += stride) p[i] = 0;
}

// ----------------------------------------------------------- degree count
__global__ void gs_count(const int* __restrict__ dst, int* __restrict__ cnt, int n) {
    int i = blockIdx.x * blockDim.x + threadIdx.x;
    if (i < n) atomicAdd(&cnt[dst[i]], 1);
}

// ------------------------------------------- block-wise exclusive scan (512)
__global__ void gs_scan(const int* __restrict__ in, int* __restrict__ out,
                        int* __restrict__ bsums, int n) {
    __shared__ int sm[SCAN_BS];
    int i = blockIdx.x * SCAN_BS + threadIdx.x;
    int v = (i < n) ? in[i] : 0;
    sm[threadIdx.x] = v;
    __syncthreads();
    for (int off = 1; off < SCAN_BS; off <<= 1) {
        int t = (threadIdx.x >= (unsigned)off) ? sm[threadIdx.x - off] : 0;
        __syncthreads();
        sm[threadIdx.x] += t;
        __syncthreads();
    }
    if (i < n) out[i] = sm[threadIdx.x] - v;            // exclusive
    if (bsums != nullptr && threadIdx.x == SCAN_BS - 1)
        bsums[blockIdx.x] = sm[SCAN_BS - 1];            // block total
}

__global__ void gs_scan_add(int* __restrict__ data, const int* __restrict__ bsums, int n) {
    int i = blockIdx.x * SCAN_BS + threadIdx.x;
    if (i < n) data[i] += bsums[blockIdx.x];
}

// -------------------------------------------------------------- CSR fill
// After this, rowptr[d] == original_start[d] + cnt[d]; aggregation recovers
// start = rowptr[d] - cnt[d], so no extra cursor array is needed.
__global__ void gs_fill(const int* __restrict__ src, const int* __restrict__ dst,
                        int* __restrict__ rowptr, int* __restrict__ nbr, int n) {
    int i = blockIdx.x * blockDim.x + threadIdx.x;
    if (i < n) {
        int d = dst[i];
        int p = atomicAdd(&rowptr[d], 1);
        nbr[p] = src[i];
    }
}

// ------------------------------------------------ gather-based mean aggregate
// One wave per node. Neighbor ids read coalesced 32-at-a-time, broadcast via
// shuffle; each lane gathers a float4 so the wave pulls the full 512B row.
// No f32 atomics anywhere. Mean normalization fused here. All feature arrays
// are L2-resident (51MB each vs 192MB L2) so this streams at L2 bandwidth.
__global__ void gs_aggregate(const float* __restrict__ xin,
                             const int* __restrict__ nbr,
                             const int* __restrict__ rowptr,
                             const int* __restrict__ cnt,
                             float* __restrict__ agg, int nNodes) {
    int node = (int)((blockIdx.x * blockDim.x + threadIdx.x) >> 5);
    int lane = threadIdx.x & 31;
    if (node >= nNodes) return;
    int deg = cnt[node];
    int start = rowptr[node] - deg;
    float4 acc = make_float4(0.f, 0.f, 0.f, 0.f);
    for (int j0 = 0; j0 < deg; j0 += 32) {
        int idx = 0;
        if (j0 + lane < deg) idx = nbr[start + j0 + lane];
        int m = min(32, deg - j0);
        for (int i = 0; i < m; ++i) {
            int s = __shfl(idx, i, 32);
            const float4 v = *(const float4*)(xin + (long long)s * D + lane * 4);
            acc.x += v.x; acc.y += v.y; acc.z += v.z; acc.w += v.w;
        }
    }
    float inv = 1.0f / fmaxf((float)deg, 1.0f);
    float4 r = make_float4(acc.x * inv, acc.y * inv, acc.z * inv, acc.w * inv);
    *(float4*)(agg + (long long)node * D + lane * 4) = r;
}

// -------------------------------------------------------- fused SAGE + WMMA
// out[m,:] = relu( agg[m,:] @ Wl + b + xin[m,:] @ Wr )
// One wave per 16-row strip computes all 128 output columns (8 tiles):
// A fragments loaded once per K-step and reused by 8 WMMAs per matrix.
// Weights staged in LDS PAIR-INTERLEAVED:
//   lds2[(k/2)*D + n] = float2{ W[k][n], W[k+1][n] }   (k even)
// so each B fragment (which always needs the even/odd K pair for a lane) is
// ONE aligned ds_load_b64 instead of two strided ds_load_b32.
__global__ void gs_sage_wmma(const float* __restrict__ xin,
                             const float* __restrict__ agg,
                             const float* __restrict__ Wl,
                             const float* __restrict__ bias,
                             const float* __restrict__ Wr,
                             float* __restrict__ out, int nNodes) {
    extern __shared__ float2 ldsW[];     // [0, 8192): Wl pairs, [8192, 16384): Wr pairs

    // cooperative pair-interleave stage: (D/2)*(D/4) float4-pair units per matrix
    for (int i = threadIdx.x; i < (D / 2) * (D / 4); i += blockDim.x) {
        int p  = i >> 5;                 // K-pair index: rows 2p, 2p+1
        int n4 = (i & 31) << 2;          // starting column of a float4
        float4 e = *(const float4*)(Wl + (2 * p)     * D + n4);
        float4 o = *(const float4*)(Wl + (2 * p + 1) * D + n4);
        float2* dL = ldsW + p * D + n4;
        dL[0] = make_float2(e.x, o.x);
        dL[1] = make_float2(e.y, o.y);
        dL[2] = make_float2(e.z, o.z);
        dL[3] = make_float2(e.w, o.w);
        float4 e2 = *(const float4*)(Wr + (2 * p)     * D + n4);
        float4 o2 = *(const float4*)(Wr + (2 * p + 1) * D + n4);
        float2* dR = ldsW + (D / 2) * D + p * D + n4;
        dR[0] = make_float2(e2.x, o2.x);
        dR[1] = make_float2(e2.y, o2.y);
        dR[2] = make_float2(e2.z, o2.z);
        dR[3] = make_float2(e2.w, o2.w);
    }
    __syncthreads();

    const int wid = (int)(blockIdx.x * (blockDim.x >> 5) + (threadIdx.x >> 5));
    const int mstrips = (nNodes + 15) >> 4;
    if (wid >= mstrips) return;          // wave-uniform: EXEC all-1s for WMMA

    const int m0 = wid << 4;
    const int lane = threadIdx.x & 31;
    const int l15 = lane & 15;
    const int half = lane >> 4;

    int row = m0 + l15;
    int rowc = row < nNodes ? row : nNodes - 1;   // clamp loads; stores guarded
    const float* arow = agg + (long long)rowc * D;
    const float* xrow = xin + (long long)rowc * D;

    v8f c[8];
#pragma unroll
    for (int t = 0; t < 8; ++t) c[t] = (v8f){0.f,0.f,0.f,0.f,0.f,0.f,0.f,0.f};

    for (int k = 0; k < D; k += 4) {
        const int kk = k + half * 2;     // this lane-half's even K base
        float2 av = *(const float2*)(arow + kk);
        float2 xv = *(const float2*)(xrow + kk);
        v2f aA; aA.x = av.x; aA.y = av.y;
        v2f aX; aX.x = xv.x; aX.y = xv.y;
        const float2* pl = ldsW + (kk >> 1) * D;               // Wl pair row
        const float2* pr = pl + (D / 2) * D;                   // Wr pair row
#pragma unroll
        for (int t = 0; t < 8; ++t) {
            const int nc = t * 16 + l15;
            float2 bl = pl[nc];                                // one ds_load_b64
            v2f b;  b.x  = bl.x;  b.y  = bl.y;
            c[t] = __builtin_amdgcn_wmma_f32_16x16x4_f32(false, aA, false, b,
                                                         (short)0, c[t], false, false);
            float2 br = pr[nc];                                // one ds_load_b64
            v2f b2; b2.x = br.x;  b2.y = br.y;
            c[t] = __builtin_amdgcn_wmma_f32_16x16x4_f32(false, aX, false, b2,
                                                         (short)0, c[t], false, false);
        }
    }

#pragma unroll
    for (int t = 0; t < 8; ++t) {
        const int nc = t * 16 + l15;
        const float bv = bias[nc];
#pragma unroll
        for (int i = 0; i < 8; ++i) {
            int r = m0 + i + half * 8;   // C layout: VGPR i -> M=i / M=8+i
            if (r < nNodes)
                out[(long long)r * D + nc] = fmaxf(c[t][i] + bv, 0.0f);
        }
    }
}

// --------------------------------------------------------------------- launch
extern "C" void kernel_launch(void* const* d_in, const int* in_sizes, int n_in,
                              void* d_out, int out_size, void* d_ws, size_t ws_size,
                              hipStream_t stream) {
    const float* x   = (const float*)d_in[0];
    const int*   ei  = (const int*)d_in[1];
    const float* W1l = (const float*)d_in[2];
    const float* b1  = (const float*)d_in[3];
    const float* W1r = (const float*)d_in[4];
    const float* W2l = (const float*)d_in[5];
    const float* b2  = (const float*)d_in[6];
    const float* W2r = (const float*)d_in[7];

    const int N = in_sizes[0] / D;
    const int E = in_sizes[1] / 2;
    const int* src = ei;
    const int* dst = ei + E;

    // workspace carve-up
    int*   cnt    = (int*)d_ws;                   // N
    int*   rowptr = cnt + N;                      // N
    int*   bsums  = rowptr + N;                   // 512
    int*   nbr    = bsums + 512;                  // E
    float* agg    = (float*)(nbr + E);            // N*D
    float* h1     = agg + (size_t)N * D;          // N*D
    float* out    = (float*)d_out;

    const int eb   = (E + 255) / 256;             // edge-parallel blocks
    const int nb   = (N + SCAN_BS - 1) / SCAN_BS; // scan blocks
    const int ab   = (N + 7) / 8;                 // aggregate: 1 wave/node, 8/block
    const int mstrips = (N + 15) / 16;
    const int sb   = (mstrips + 7) / 8;           // sage: 1 wave/strip, 8/block
    const size_t wbytes = 2 * D * D * sizeof(float); // 128KB dynamic LDS

    // ---- CSR build (once; reused by both layers) ----
    gs_zero_i32<<<1024, 256, 0, stream>>>(cnt, N);
    gs_count<<<eb, 256, 0, stream>>>(dst, cnt, E);
    gs_scan<<<nb, SCAN_BS, 0, stream>>>(cnt, rowptr, bsums, N);
    gs_scan<<<1, SCAN_BS, 0, stream>>>(bsums, bsums, nullptr, nb);
    gs_scan_add<<<nb, SCAN_BS, 0, stream>>>(rowptr, bsums, N);
    gs_fill<<<eb, 256, 0, stream>>>(src, dst, rowptr, nbr, E);

    // ---- layer 1 ----
    gs_aggregate<<<ab, 256, 0, stream>>>(x, nbr, rowptr, cnt, agg, N);
    gs_sage_wmma<<<sb, 256, wbytes, stream>>>(x, agg, W1l, b1, W1r, h1, N);

    // ---- layer 2 ----
    gs_aggregate<<<ab, 256, 0, stream>>>(h1, nbr, rowptr, cnt, agg, N);
    gs_sage_wmma<<<sb, 256, wbytes, stream>>>(h1, agg, W2l, b2, W2r, out, N);
}